// EctLayer_1769526526454
// MI455X (gfx1250) — compile-verified
//
#include <hip/hip_runtime.h>
#include <cstdint>
#include <cstddef>

// ECT layer, fused: nh = x@dirs via V_WMMA_F32_16X16X4_F32 (K=3 padded to 4),
// sigmoid + segment-sum in registers. Output ownership is exclusive per thread
// -> no atomics, deterministic.
//
// sigmoid(500*(lin - h)) = 0.5 + 0.5*tanh(250*(lin - h))  -> v_tanh_f32 (gfx1250)
// accumulated as raw tanh, affine-corrected once per segment.
//
// Shapes: x[N][3], dirs[3][64], lin[64], batch[N] (int64, sorted, in [0,B)),
// out[B][64][64].

typedef __attribute__((ext_vector_type(2))) float v2f;
typedef __attribute__((ext_vector_type(8))) float v8f;

#define T_DIM 64
#define LOG2E 1.44269504088896340736f
#define SCALE_F 500.0f

#if defined(__has_builtin)
#if __has_builtin(__builtin_amdgcn_tanhf)
#define HAVE_TANH 1
#endif
#endif
#ifndef HAVE_TANH
#define HAVE_TANH 0
#endif

__global__ __launch_bounds__(256) void ect_fused_kernel(
    const float* __restrict__ x,          // [N,3]
    const float* __restrict__ dirs,       // [3,64]
    const float* __restrict__ lin,        // [R]
    const long long* __restrict__ batch,  // [N], sorted
    float* __restrict__ out,              // [B,R,64]
    int N, int R)
{
    const int b     = blockIdx.x;   // segment id
    const int rtile = blockIdx.y;   // 8 r-values per tile
    const int tid   = threadIdx.x;  // 0..255
    const int lane  = tid & 31;
    const int wave  = tid >> 5;     // 0..7

    __shared__ float nh[32][T_DIM + 1];  // chunk of 32 points' heights (padded)
    __shared__ int   seg[2];

    // ---- segment bounds: lower_bound(batch, b) and lower_bound(batch, b+1) ----
    if (tid < 2) {
        long long key = (long long)(b + tid);
        int lo = 0, hi = N;
        while (lo < hi) {
            int mid = (lo + hi) >> 1;
            if (batch[mid] < key) lo = mid + 1; else hi = mid;
        }
        seg[tid] = lo;
    }
    __syncthreads();
    const int s0 = seg[0], s1 = seg[1];

    // ---- per-wave WMMA B-matrix: directions tile (4 x 16), K padded with 0 ----
    // wave w owns nh sub-tile rows [ (w>>2)*16, +16 ) x cols [ (w&3)*16, +16 )
    const int ttile = (wave & 3) * 16;
    const int ntile = (wave >> 2) * 16;
    const bool lo_half = (lane < 16);
    v2f bmat;
    {
        const int col = ttile + (lane & 15);
        bmat.x = lo_half ? dirs[0 * T_DIM + col] : dirs[2 * T_DIM + col];
        bmat.y = lo_half ? dirs[1 * T_DIM + col] : 0.0f;
    }

    // ---- per-thread accumulators: t = tid%64, r = rtile*8 + (tid/64)*2 + {0,1} ----
    const int t_own = tid & 63;
    const int r0    = rtile * 8 + (tid >> 6) * 2;

#if HAVE_TANH
    const float C = 0.5f * SCALE_F;              // 250: tanh argument scale
#else
    const float C = SCALE_F * LOG2E;             // fold SCALE and log2(e) for exp2
#endif
    const float a0 = C * lin[r0 + 0];
    const float a1 = C * lin[r0 + 1];
    float acc0 = 0.0f, acc1 = 0.0f;

    for (int base = s0; base < s1; base += 32) {
        // ---- nh[0..31][0..63] for points base..base+31 via WMMA ----
        {
            int row = base + ntile + (lane & 15);
            if (row >= N) row = N - 1;           // clamp; masked out of the sum below
            // branchless contiguous 3-dword read, then lane-select (no exec games)
            const float x0 = x[row * 3 + 0];
            const float x1 = x[row * 3 + 1];
            const float x2 = x[row * 3 + 2];
            v2f amat;
            amat.x = lo_half ? x0 : x2;          // K0 | K2
            amat.y = lo_half ? x1 : 0.0f;        // K1 | K3(pad)
            v8f c = {};
            c = __builtin_amdgcn_wmma_f32_16x16x4_f32(
                    /*neg_a=*/false, amat, /*neg_b=*/false, bmat,
                    /*c_mod=*/(short)0, c, /*reuse_a=*/false, /*reuse_b=*/false);
            // D layout: lanes 0-15 hold rows M=v (v=0..7), lanes 16-31 rows M=8+v
            const int mbase = ntile + (lo_half ? 0 : 8);
            const int col   = ttile + (lane & 15);
            #pragma unroll
            for (int v = 0; v < 8; ++v)
                nh[mbase + v][col] = c[v];
        }
        __syncthreads();

        int lim = s1 - base; if (lim > 32) lim = 32;
#if HAVE_TANH
        // acc += tanh(C*h - a_j); sigmoid recovered as 0.5*count - 0.5*acc at end.
        // note sigmoid(a - Ch form): sigmoid(S*(lin-h)) = 0.5 - 0.5*tanh(C*h - C*lin)
        for (int nn = 0; nn < lim; ++nn) {
            const float h = nh[nn][t_own];
            acc0 += __builtin_amdgcn_tanhf(__builtin_fmaf(C, h, -a0));
            acc1 += __builtin_amdgcn_tanhf(__builtin_fmaf(C, h, -a1));
        }
#else
        // sigmoid = rcp(1 + exp2(C*h - C*lin))
        for (int nn = 0; nn < lim; ++nn) {
            const float h  = nh[nn][t_own];
            const float e0 = __builtin_amdgcn_exp2f(__builtin_fmaf(C, h, -a0));
            const float e1 = __builtin_amdgcn_exp2f(__builtin_fmaf(C, h, -a1));
            acc0 += __builtin_amdgcn_rcpf(1.0f + e0);
            acc1 += __builtin_amdgcn_rcpf(1.0f + e1);
        }
#endif
        __syncthreads();
    }

#if HAVE_TANH
    // sigmoid sum = 0.5*count - 0.5*sum(tanh)
    const float half_cnt = 0.5f * (float)(s1 - s0);
    acc0 = __builtin_fmaf(-0.5f, acc0, half_cnt);
    acc1 = __builtin_fmaf(-0.5f, acc1, half_cnt);
#endif

    // ---- exclusive writes: out[b][r0..r0+1][t_own] ----
    float* o = out + ((size_t)b * R + r0) * T_DIM + t_own;
    o[0]     = acc0;
    o[T_DIM] = acc1;
}

extern "C" void kernel_launch(void* const* d_in, const int* in_sizes, int n_in,
                              void* d_out, int out_size, void* d_ws, size_t ws_size,
                              hipStream_t stream) {
    (void)n_in; (void)d_ws; (void)ws_size;
    const float*     x     = (const float*)d_in[0];
    const float*     dirs  = (const float*)d_in[1];
    const float*     lin   = (const float*)d_in[2];
    const long long* batch = (const long long*)d_in[3];   // int64 per reference
    float*           out   = (float*)d_out;

    const int N = in_sizes[3];           // 65536 points
    const int R = in_sizes[2];           // 64 thresholds
    const int T = in_sizes[1] / 3;       // 64 directions (kernel assumes 64)
    const int B = out_size / (R * T);    // 64 segments

    dim3 grid(B, R / 8);                 // 64 x 8 = 512 blocks, exclusive outputs
    ect_fused_kernel<<<grid, 256, 0, stream>>>(x, dirs, lin, batch, out, N, R);
}